// ComplexGCNModel_29901562315006
// MI455X (gfx1250) — compile-verified
//
#include <hip/hip_runtime.h>
#include <stdint.h>

typedef __attribute__((ext_vector_type(2))) float v2f;
typedef __attribute__((ext_vector_type(8))) float v8f;

#define THREADS 256

// ---------------------------------------------------------------------------
// Degree / normalization precompute (shared by all 4 layers)
// ---------------------------------------------------------------------------
__global__ void deg_init_kernel(float* __restrict__ deg, int n) {
    unsigned i = blockIdx.x * blockDim.x + threadIdx.x;
    if (i < (unsigned)n) deg[i] = 1.0f;  // self-loop: deg = in-degree + 1
}

__global__ void deg_count_kernel(const long long* __restrict__ dst,
                                 float* __restrict__ deg, int e) {
    unsigned i = blockIdx.x * blockDim.x + threadIdx.x;
    if (i < (unsigned)e) unsafeAtomicAdd(&deg[(int)dst[i]], 1.0f);
}

__global__ void dinv_kernel(const float* __restrict__ deg,
                            float* __restrict__ dinv,
                            float* __restrict__ dinv2, int n) {
    unsigned i = blockIdx.x * blockDim.x + threadIdx.x;
    if (i < (unsigned)n) {
        float r = rsqrtf(deg[i]);
        dinv[i]  = r;
        dinv2[i] = r * r;
    }
}

// Edge prep: int32 endpoints + per-edge symmetric norm, computed once.
__global__ void edge_prep_kernel(const long long* __restrict__ src,
                                 const long long* __restrict__ dst,
                                 const float* __restrict__ dinv,
                                 int* __restrict__ srcI, int* __restrict__ dstI,
                                 float* __restrict__ enorm, int e) {
    unsigned i = blockIdx.x * blockDim.x + threadIdx.x;
    if (i < (unsigned)e) {
        int s = (int)src[i];
        int d = (int)dst[i];
        srcI[i]  = s;
        dstI[i]  = d;
        enorm[i] = dinv[s] * dinv[d];
    }
}

// ---------------------------------------------------------------------------
// W transpose (+ K zero-pad): Wt[n*Kp + k] = (k<K) ? W[k*N + n] : 0
// ---------------------------------------------------------------------------
__global__ void wt_kernel(const float* __restrict__ W, float* __restrict__ Wt,
                          int K, int N, int Kp, unsigned total) {
    unsigned i = blockIdx.x * blockDim.x + threadIdx.x;
    if (i < total) {
        unsigned n = i / (unsigned)Kp;
        unsigned k = i % (unsigned)Kp;
        Wt[i] = (k < (unsigned)K) ? W[(size_t)k * N + n] : 0.f;
    }
}

// X K-pad (layer 1: K=2 -> 4): Xp[row*Kp + k] = (k<K) ? X[row*K + k] : 0
__global__ void xpad_kernel(const float* __restrict__ X, float* __restrict__ Xp,
                            int K, int Kp, unsigned total) {
    unsigned i = blockIdx.x * blockDim.x + threadIdx.x;
    if (i < total) {
        unsigned row = i / (unsigned)Kp;
        unsigned k   = i % (unsigned)Kp;
        Xp[i] = (k < (unsigned)K) ? X[(size_t)row * K + k] : 0.f;
    }
}

// ---------------------------------------------------------------------------
// Fast fp32 WMMA GEMM: H[M,N] = relu?(X[M,K]) @ W[K,N], W pre-transposed (Wt[N,K]).
// Requires M%16==0, K%4==0, N%64==0. One 16x64 tile per wave (4 accumulators).
// No per-lane guards -> EXEC all-ones at every WMMA; all frag loads are b64.
// A frag: lane l holds X[tm*16 + (l&15)][kb], X[...][kb+1], kb = k0+(l>>4)*2
// B frag: lane l holds Wt[col][kb], Wt[col][kb+1],          col = ctile*16+(l&15)
// ---------------------------------------------------------------------------
template <bool RELU>
__global__ void gcn_gemm_wmma_fast(const float* __restrict__ X,
                                   const float* __restrict__ Wt,
                                   float* __restrict__ H,
                                   int K, int N, int tilesN64, int numWaves) {
    int wave = (int)((blockIdx.x * blockDim.x + threadIdx.x) >> 5);
    if (wave >= numWaves) return;            // wave-uniform exit
    const int lane   = threadIdx.x & 31;
    const int lane16 = lane & 15;
    const int half   = lane >> 4;
    const int tm = wave / tilesN64;
    const int tn = wave % tilesN64;

    const float* ap = X + (size_t)(tm * 16 + lane16) * K + half * 2;
    const int colBase = tn * 64 + lane16;
    const float* bp0 = Wt + (size_t)(colBase)      * K + half * 2;
    const float* bp1 = Wt + (size_t)(colBase + 16) * K + half * 2;
    const float* bp2 = Wt + (size_t)(colBase + 32) * K + half * 2;
    const float* bp3 = Wt + (size_t)(colBase + 48) * K + half * 2;

    v8f acc0 = {0.f,0.f,0.f,0.f,0.f,0.f,0.f,0.f};
    v8f acc1 = acc0, acc2 = acc0, acc3 = acc0;

#pragma unroll 2
    for (int k0 = 0; k0 < K; k0 += 4) {
        v2f a  = *(const v2f*)(ap  + k0);
        v2f b0 = *(const v2f*)(bp0 + k0);
        v2f b1 = *(const v2f*)(bp1 + k0);
        v2f b2 = *(const v2f*)(bp2 + k0);
        v2f b3 = *(const v2f*)(bp3 + k0);
        if (RELU) {
            a.x = fmaxf(a.x, 0.f);
            a.y = fmaxf(a.y, 0.f);
        }
        acc0 = __builtin_amdgcn_wmma_f32_16x16x4_f32(false, a, false, b0,
                                                     (short)0, acc0, false, false);
        acc1 = __builtin_amdgcn_wmma_f32_16x16x4_f32(false, a, false, b1,
                                                     (short)0, acc1, false, false);
        acc2 = __builtin_amdgcn_wmma_f32_16x16x4_f32(false, a, false, b2,
                                                     (short)0, acc2, false, false);
        acc3 = __builtin_amdgcn_wmma_f32_16x16x4_f32(false, a, false, b3,
                                                     (short)0, acc3, false, false);
    }

    // C/D layout: vgpr v -> row tm*16 + v + 8*half, col = ctile*16 + lane16
    float* hp = H + (size_t)(tm * 16 + 8 * half) * N + tn * 64 + lane16;
#pragma unroll
    for (int v = 0; v < 8; ++v) {
        float* r = hp + (size_t)v * N;
        r[0]  = acc0[v];
        r[16] = acc1[v];
        r[32] = acc2[v];
        r[48] = acc3[v];
    }
}

// ---------------------------------------------------------------------------
// Small-N GEMM (final layer, N=C<=4): one thread per node, float4 K-loop.
// ---------------------------------------------------------------------------
__global__ void gemm_smallN_kernel(const float* __restrict__ X,
                                   const float* __restrict__ W,
                                   float* __restrict__ H,
                                   int M, int K, int C) {
    unsigned v = blockIdx.x * blockDim.x + threadIdx.x;
    if (v >= (unsigned)M) return;
    const float* xr = X + (size_t)v * K;
    float acc[4] = {0.f, 0.f, 0.f, 0.f};
    for (int k = 0; k < K; k += 4) {
        float4 xv = *(const float4*)(xr + k);
        xv.x = fmaxf(xv.x, 0.f);  xv.y = fmaxf(xv.y, 0.f);
        xv.z = fmaxf(xv.z, 0.f);  xv.w = fmaxf(xv.w, 0.f);
        for (int c = 0; c < C; ++c) {   // W rows are tiny & uniform -> cached
            acc[c] += xv.x * W[(size_t)(k + 0) * C + c]
                    + xv.y * W[(size_t)(k + 1) * C + c]
                    + xv.z * W[(size_t)(k + 2) * C + c]
                    + xv.w * W[(size_t)(k + 3) * C + c];
        }
    }
    for (int c = 0; c < C; ++c) H[(size_t)v * C + c] = acc[c];
}

// ---------------------------------------------------------------------------
// AGG = H * dinv2[v] + b[c]   (self-loop + bias; initializes AGG)
// ---------------------------------------------------------------------------
__global__ void agg_init_vec4_kernel(const float* __restrict__ H,
                                     const float* __restrict__ dinv2,
                                     const float* __restrict__ bias,
                                     float* __restrict__ AGG,
                                     unsigned total4, int C4) {
    unsigned i = blockIdx.x * blockDim.x + threadIdx.x;
    if (i >= total4) return;
    unsigned v = i / (unsigned)C4;
    unsigned c = (i % (unsigned)C4) * 4u;
    float s = dinv2[v];
    float4 h = *(const float4*)(H + (size_t)i * 4);
    const float4 b = *(const float4*)(bias + c);
    float4 o;
    o.x = h.x * s + b.x;  o.y = h.y * s + b.y;
    o.z = h.z * s + b.z;  o.w = h.w * s + b.w;
    *(float4*)(AGG + (size_t)i * 4) = o;
}

__global__ void agg_init_scalar_kernel(const float* __restrict__ H,
                                       const float* __restrict__ dinv2,
                                       const float* __restrict__ bias,
                                       float* __restrict__ AGG,
                                       unsigned total, int C) {
    unsigned i = blockIdx.x * blockDim.x + threadIdx.x;
    if (i < total)
        AGG[i] = H[i] * dinv2[i / (unsigned)C] + bias[i % (unsigned)C];
}

// ---------------------------------------------------------------------------
// Edge scatter: AGG[dst] += enorm[e] * H[src]
// One thread per (edge, 4-channel chunk): float4 gather + 4 hw float atomics.
// ---------------------------------------------------------------------------
__global__ void scatter_vec4_kernel(const float* __restrict__ H,
                                    const int* __restrict__ srcI,
                                    const int* __restrict__ dstI,
                                    const float* __restrict__ enorm,
                                    float* __restrict__ AGG,
                                    unsigned total, int C4, int C) {
    unsigned tid = blockIdx.x * blockDim.x + threadIdx.x;
    if (tid >= total) return;
    unsigned e = tid / (unsigned)C4;
    unsigned c = (tid % (unsigned)C4) * 4u;
    int s = srcI[e];
    int d = dstI[e];
    float nrm = enorm[e];
    float4 h = *(const float4*)(H + (size_t)s * C + c);
    float* ap = AGG + (size_t)d * C + c;
    unsafeAtomicAdd(ap + 0, h.x * nrm);
    unsafeAtomicAdd(ap + 1, h.y * nrm);
    unsafeAtomicAdd(ap + 2, h.z * nrm);
    unsafeAtomicAdd(ap + 3, h.w * nrm);
}

__global__ void scatter_scalar_kernel(const float* __restrict__ H,
                                      const int* __restrict__ srcI,
                                      const int* __restrict__ dstI,
                                      const float* __restrict__ enorm,
                                      float* __restrict__ AGG,
                                      unsigned total, int C) {
    unsigned tid = blockIdx.x * blockDim.x + threadIdx.x;
    if (tid >= total) return;
    unsigned e = tid / (unsigned)C;
    unsigned c = tid % (unsigned)C;
    unsafeAtomicAdd(AGG + (size_t)dstI[e] * C + c,
                    H[(size_t)srcI[e] * C + c] * enorm[e]);
}

// ---------------------------------------------------------------------------
static inline unsigned blocks_for(size_t total) {
    return (unsigned)((total + THREADS - 1) / THREADS);
}

extern "C" void kernel_launch(void* const* d_in, const int* in_sizes, int n_in,
                              void* d_out, int out_size, void* d_ws, size_t ws_size,
                              hipStream_t stream) {
    (void)n_in; (void)out_size; (void)ws_size;

    const float*     x  = (const float*)d_in[0];
    const long long* ei = (const long long*)d_in[1];   // int64 edge_index [2,E]
    const float* Wm[4]  = {(const float*)d_in[2], (const float*)d_in[4],
                           (const float*)d_in[6], (const float*)d_in[8]};
    const float* bv[4]  = {(const float*)d_in[3], (const float*)d_in[5],
                           (const float*)d_in[7], (const float*)d_in[9]};

    int dims[5];
    dims[1] = in_sizes[3];
    dims[2] = in_sizes[5];
    dims[3] = in_sizes[7];
    dims[4] = in_sizes[9];
    dims[0] = in_sizes[2] / dims[1];
    const int N = in_sizes[0] / dims[0];
    const int E = in_sizes[1] / 2;
    const long long* srcp = ei;
    const long long* dstp = ei + E;

    int maxC = dims[1], maxKN = 0;
    for (int i = 1; i <= 4; ++i) if (dims[i] > maxC) maxC = dims[i];
    for (int i = 0; i < 4; ++i) {
        int kp = dims[i] < 4 ? 4 : dims[i];
        if (kp * dims[i + 1] > maxKN) maxKN = kp * dims[i + 1];
    }

    // Workspace: deg|dinv|dinv2|srcI|dstI|enorm|Xp|Wt|H|AGG
    auto align256 = [](size_t v) { return (v + 255) & ~(size_t)255; };
    char*  p     = (char*)d_ws;
    auto carve = [&](size_t bytes) { char* r = p; p += align256(bytes); return r; };
    float* deg   = (float*)carve((size_t)N * 4);
    float* dinv  = (float*)carve((size_t)N * 4);
    float* dinv2 = (float*)carve((size_t)N * 4);
    int*   srcI  = (int*)  carve((size_t)E * 4);
    int*   dstI  = (int*)  carve((size_t)E * 4);
    float* enorm = (float*)carve((size_t)E * 4);
    float* Xp    = (float*)carve((size_t)N * 4 * 4);       // K-pad staging (K<4)
    float* Wt    = (float*)carve((size_t)maxKN * 4);       // transposed weights
    float* Hbuf  = (float*)carve((size_t)N * maxC * 4);
    float* Abuf  = (float*)carve((size_t)N * maxC * 4);

    // --- normalization + edge precompute ---
    deg_init_kernel<<<blocks_for(N), THREADS, 0, stream>>>(deg, N);
    deg_count_kernel<<<blocks_for(E), THREADS, 0, stream>>>(dstp, deg, E);
    dinv_kernel<<<blocks_for(N), THREADS, 0, stream>>>(deg, dinv, dinv2, N);
    edge_prep_kernel<<<blocks_for(E), THREADS, 0, stream>>>(
        srcp, dstp, dinv, srcI, dstI, enorm, E);

    // --- 4 GCN layers ---
    const float* layer_in = x;
    for (int l = 0; l < 4; ++l) {
        const int K = dims[l];
        const int C = dims[l + 1];
        float* agg = (l == 3) ? (float*)d_out : Abuf;
        const bool relu_in = (l > 0);

        if (C % 64 == 0) {
            // Fast WMMA path (layers 1..3). Pad K to multiple of 4 if needed.
            int Kp = (K + 3) & ~3;
            const float* Xin = layer_in;
            if (Kp != K) {
                unsigned tot = (unsigned)N * (unsigned)Kp;
                xpad_kernel<<<blocks_for(tot), THREADS, 0, stream>>>(
                    layer_in, Xp, K, Kp, tot);
                Xin = Xp;
                // note: relu_in is false for layer 0 (the only padded layer)
            }
            unsigned wtot = (unsigned)C * (unsigned)Kp;
            wt_kernel<<<blocks_for(wtot), THREADS, 0, stream>>>(
                Wm[l], Wt, K, C, Kp, wtot);

            int tilesN64 = C / 64;
            int numWaves = (N / 16) * tilesN64;
            int blocks   = (numWaves + 7) / 8;
            if (relu_in)
                gcn_gemm_wmma_fast<true><<<blocks, THREADS, 0, stream>>>(
                    Xin, Wt, Hbuf, Kp, C, tilesN64, numWaves);
            else
                gcn_gemm_wmma_fast<false><<<blocks, THREADS, 0, stream>>>(
                    Xin, Wt, Hbuf, Kp, C, tilesN64, numWaves);
        } else {
            // Tiny-N path (final layer): per-node VALU dot products.
            gemm_smallN_kernel<<<blocks_for(N), THREADS, 0, stream>>>(
                layer_in, Wm[l], Hbuf, N, K, C);
        }

        // AGG = H * dinv2 + b
        if ((C & 3) == 0) {
            unsigned tot4 = (unsigned)N * (unsigned)(C / 4);
            agg_init_vec4_kernel<<<blocks_for(tot4), THREADS, 0, stream>>>(
                Hbuf, dinv2, bv[l], agg, tot4, C / 4);
        } else {
            unsigned tot = (unsigned)N * (unsigned)C;
            agg_init_scalar_kernel<<<blocks_for(tot), THREADS, 0, stream>>>(
                Hbuf, dinv2, bv[l], agg, tot, C);
        }

        // AGG[dst] += enorm * H[src]
        if ((C & 3) == 0) {
            int C4 = C >> 2;
            unsigned totalEC = (unsigned)E * (unsigned)C4;
            scatter_vec4_kernel<<<blocks_for(totalEC), THREADS, 0, stream>>>(
                Hbuf, srcI, dstI, enorm, agg, totalEC, C4, C);
        } else {
            unsigned totalEC = (unsigned)E * (unsigned)C;
            scatter_scalar_kernel<<<blocks_for(totalEC), THREADS, 0, stream>>>(
                Hbuf, srcI, dstI, enorm, agg, totalEC, C);
        }

        layer_in = agg;   // next layer reads AGG (ReLU fused at GEMM load)
    }
}